// HungarianMatcher_48567490183395
// MI455X (gfx1250) — compile-verified
//
#include <hip/hip_runtime.h>
#include <hip/hip_bf16.h>

// HungarianMatcher for MI455X (gfx1250, wave32).
//
// outputs: [8][512][2] f32, targets: [8][256][2] f32
// d_out (float*): row_ind[8][256] ++ col_ind[8][256] ++ C[8][512][256]
//
// Kernel 1: cost matrix via V_WMMA_F32_16X16X4_F32 using
//   |dx|+|dy| = max(|dx+dy|, |dx-dy|)
// Two K=4 f32 WMMAs per 16x16 tile (same 16 targets, "+" and "-" column
// variants), so the max is lane-local VALU: no cross-lane ops, no divergence.
// Writes C and (if workspace permits) the transposed costT in the same pass.
// Kernel 2: JV shortest-augmenting-path, one workgroup (16 wave32) per batch,
//   all solver state LDS-resident, argmin via wave32 shfl_xor reduction.

typedef float v2f __attribute__((ext_vector_type(2)));
typedef float v8f __attribute__((ext_vector_type(8)));

#define NB 8
#define NQ 512
#define NT 256
#define HINF 1000000000.0f

// ---------------------------------------------------------------------------
// Kernel 1: cost matrix via WMMA f32 16x16x4.
// grid = NB*32 blocks (b = blk>>5, qtile = blk&31), 128 threads = 4 waves.
// Each wave owns 4 target-tiles of 16 targets (16 tiles cover 256 targets).
// Per tile: 2 WMMAs -> per-lane max(|acc+|,|acc-|) -> dense stores.
// ---------------------------------------------------------------------------
__global__ __launch_bounds__(128) void cost_wmma_kernel(
    const float2* __restrict__ outputs,   // [NB*NQ]
    const float2* __restrict__ targets,   // [NB*NT]
    float* __restrict__ Cout,             // [NB][NQ][NT]
    float* __restrict__ costT,            // [NB][NT][NQ] or nullptr
    int writeT)
{
    const int b     = blockIdx.x >> 5;
    const int qtile = blockIdx.x & 31;
    const int wave  = threadIdx.x >> 5;
    const int lane  = threadIdx.x & 31;
    const int half  = lane >> 4;     // K-half for A/B operands
    const int n     = lane & 15;     // A row (M) / B column (N)

    // A tile 16x4: row m -> (ox, oy, 1, 0).  Layout: VGPR0 holds K=0
    // (lanes 0-15) / K=2 (lanes 16-31); VGPR1 holds K=1 / K=3.
    const int q = qtile * 16 + n;
    const float2 o = outputs[b * NQ + q];
    v2f a;
    a.x = half ? 1.0f : o.x;   // K=0 row: ox ; K=2 row: 1
    a.y = half ? 0.0f : o.y;   // K=1 row: oy ; K=3 row: 0

    for (int it = 0; it < 4; ++it) {
        const int ttile = wave + (it << 2);   // 0..15
        const int t0    = ttile * 16;
        const int t     = t0 + n;             // this lane's target column
        const float2 tg = targets[b * NT + t];

        // B "+" tile: rows (1, 1, -(tx+ty), 0)  ->  D = dx + dy
        v2f bp;
        bp.x = half ? (-(tg.x + tg.y)) : 1.0f;
        bp.y = half ? 0.0f : 1.0f;
        // B "-" tile: rows (1, -1, -(tx-ty), 0) ->  D = dx - dy
        v2f bm;
        bm.x = half ? (-(tg.x - tg.y)) : 1.0f;
        bm.y = half ? 0.0f : -1.0f;

        v8f accp = {0.f, 0.f, 0.f, 0.f, 0.f, 0.f, 0.f, 0.f};
        v8f accm = {0.f, 0.f, 0.f, 0.f, 0.f, 0.f, 0.f, 0.f};
        accp = __builtin_amdgcn_wmma_f32_16x16x4_f32(
            false, a, false, bp, (short)0, accp, false, false);
        accm = __builtin_amdgcn_wmma_f32_16x16x4_f32(
            false, a, false, bm, (short)0, accm, false, false);

        // C/D layout: VGPR r, lanes 0-15 -> M=r, lanes 16-31 -> M=r+8.
        // |dx|+|dy| = max(|dx+dy|, |dx-dy|): lane-local, no shuffles.
        #pragma unroll
        for (int r = 0; r < 8; ++r) {
            const float vv = fmaxf(fabsf(accp[r]), fabsf(accm[r]));
            const int qq = qtile * 16 + r + (half ? 8 : 0);
            Cout[(b * NQ + qq) * NT + t] = vv;
            if (writeT) costT[((size_t)b * NT + t) * NQ + qq] = vv;
        }
    }
}

// ---------------------------------------------------------------------------
// Kernel 2: Jonker-Volgenant shortest augmenting path. One block per batch,
// 512 threads (16 wave32 waves), one thread per column. Matches the JAX
// reference step-for-step (incl. first-occurrence argmin tie-breaks).
// ---------------------------------------------------------------------------
__global__ __launch_bounds__(512) void hungarian_kernel(
    const float* __restrict__ costT,  // [NB][NT][NQ] (if useT)
    const float* __restrict__ Cmat,   // [NB][NQ][NT] (fallback)
    int useT,
    float* __restrict__ rowOut,       // [NB][NT]
    float* __restrict__ colOut)       // [NB][NT]
{
    __shared__ float u[NT];
    __shared__ float v[NQ + 1];
    __shared__ int   p[NQ + 1];
    __shared__ float minv[NQ];
    __shared__ int   way[NQ];
    __shared__ int   used[NQ + 1];
    __shared__ int   scan[NQ];
    __shared__ float wval[16];
    __shared__ int   widx[16];
    __shared__ int   sj0, si0, sdone, sjmin;
    __shared__ float sdelta;

    const int b   = blockIdx.x;
    const int tid = threadIdx.x;

    if (tid < NT) u[tid] = 0.0f;
    v[tid] = 0.0f;
    p[tid] = -1;
    if (tid == 0) { v[NQ] = 0.0f; p[NQ] = -1; }
    __syncthreads();

    for (int i = 0; i < NT; ++i) {
        minv[tid] = HINF;
        way[tid]  = 0;
        used[tid] = 0;
        if (tid == 0) { used[NQ] = 0; p[NQ] = i; sj0 = NQ; }
        __syncthreads();

        while (true) {
            if (tid == 0) {
                const int j0 = sj0;
                const int pi = p[j0];
                sdone = (pi == -1);
                if (!sdone) { used[j0] = 1; si0 = pi; }
            }
            __syncthreads();
            if (sdone) break;

            const int   j0cur = sj0;
            const int   i0    = si0;
            const float ui0   = u[i0];
            const float c = useT
                ? costT[((size_t)b * NT + i0) * NQ + tid]
                : Cmat[((size_t)b * NQ + tid) * NT + i0];
            const float cur = c - ui0 - v[tid];
            const int isUsed = used[tid];
            if (!isUsed && cur < minv[tid]) { minv[tid] = cur; way[tid] = j0cur; }

            // argmin over masked minv: wave32 shfl_xor reduce, then cross-wave.
            float bv = isUsed ? HINF : minv[tid];
            int   bi = tid;
            #pragma unroll
            for (int m = 1; m < 32; m <<= 1) {
                float ov = __shfl_xor(bv, m);
                int   oi = __shfl_xor(bi, m);
                if (ov < bv || (ov == bv && oi < bi)) { bv = ov; bi = oi; }
            }
            if ((tid & 31) == 0) { wval[tid >> 5] = bv; widx[tid >> 5] = bi; }
            __syncthreads();
            if (tid == 0) {
                #pragma unroll
                for (int w = 1; w < 16; ++w) {
                    if (wval[w] < bv || (wval[w] == bv && widx[w] < bi)) {
                        bv = wval[w]; bi = widx[w];
                    }
                }
                sdelta = bv; sjmin = bi;
            }
            __syncthreads();

            const float delta = sdelta;
            if (used[tid]) { u[p[tid]] += delta; v[tid] -= delta; }
            else           { minv[tid] -= delta; }
            if (tid == 0 && used[NQ]) { u[p[NQ]] += delta; v[NQ] -= delta; }
            __syncthreads();
            if (tid == 0) sj0 = sjmin;
            __syncthreads();
        }

        // Augment along the alternating path (sequential, thread 0).
        if (tid == 0) {
            int j0 = sj0;
            while (j0 != NQ) {
                const int j1 = way[j0];
                p[j0] = p[j1];
                j0 = j1;
            }
        }
        __syncthreads();
    }

    // Emit (row_ind, col_ind): pairs (q, p[q]) for valid q, sorted by q.
    const int valid = (p[tid] >= 0) ? 1 : 0;
    scan[tid] = valid;
    __syncthreads();
    for (int off = 1; off < NQ; off <<= 1) {
        const int x = (tid >= off) ? scan[tid - off] : 0;
        __syncthreads();
        scan[tid] += x;
        __syncthreads();
    }
    if (valid) {
        const int k = scan[tid] - 1;
        rowOut[b * NT + k] = (float)tid;
        colOut[b * NT + k] = (float)p[tid];
    }
}

// ---------------------------------------------------------------------------
extern "C" void kernel_launch(void* const* d_in, const int* in_sizes, int n_in,
                              void* d_out, int out_size, void* d_ws, size_t ws_size,
                              hipStream_t stream) {
    (void)in_sizes; (void)n_in; (void)out_size;
    const float2* outputs = (const float2*)d_in[0];   // 8*512 float2
    const float2* targets = (const float2*)d_in[1];   // 8*256 float2
    float* out    = (float*)d_out;
    float* rowOut = out;                    // NB*NT
    float* colOut = out + NB * NT;          // NB*NT
    float* Cmat   = out + 2 * NB * NT;      // NB*NQ*NT

    const size_t needT = (size_t)NB * NT * NQ * sizeof(float);
    const int useT = (d_ws != nullptr && ws_size >= needT) ? 1 : 0;
    float* costT = (float*)d_ws;

    cost_wmma_kernel<<<NB * 32, 128, 0, stream>>>(outputs, targets, Cmat,
                                                  costT, useT);
    hungarian_kernel<<<NB, 512, 0, stream>>>(costT, Cmat, useT, rowOut, colOut);
}